// MoveScalePSF_12378095747577
// MI455X (gfx1250) — compile-verified
//
#include <hip/hip_runtime.h>
#include <math.h>

#define CC 16
#define FF 32
#define HH 256
#define WW 256
#define ROWS 32   // output rows per block; stages ROWS+1 psf rows in LDS

typedef unsigned int u32x4 __attribute__((ext_vector_type(4)));
typedef int          i32x8 __attribute__((ext_vector_type(8)));
typedef int          i32x4 __attribute__((ext_vector_type(4)));

// -------- Pre-kernel: per-(c,f) shift params + partial_cords tail --------
// ws layout: 2 float4 per (c,f):  [ox(bits), oy(bits), a00, a01], [a10, a11, 0, 0]
__global__ __launch_bounds__(256) void MoveScalePSF_params(
    const float* __restrict__ rot, const float* __restrict__ xin,
    const float* __restrict__ yin, const float* __restrict__ flux,
    float* __restrict__ out, float4* __restrict__ ws)
{
    const int t = blockIdx.x * 256 + threadIdx.x;   // 0..511
    if (t < CC * FF) {
        const int c = t >> 5;       // /FF
        const int f = t & (FF - 1);
        const float xc = xin[c] - 128.0f;
        const float yc = yin[c] - 128.0f;
        const float radius = sqrtf(xc * xc + yc * yc);
        float theta = fmodf(atan2f(yc, xc) * 57.295779513082320876f, 360.0f);
        if (theta < 0.0f) theta += 360.0f;
        const float rt = (theta - rot[f]) * 0.017453292519943295f;
        const float sx = -radius * cosf(rt);      // = -x_s
        const float sy = -radius * sinf(rt);      // = -y_s
        const float fox = floorf(sx), foy = floorf(sy);
        const int   ox = (int)fox,    oy = (int)foy;
        const float wx = sx - fox,    wy = sy - foy;
        const float fl  = flux[c];
        ws[2 * t]     = make_float4(__int_as_float(ox), __int_as_float(oy),
                                    fl * (1.0f - wy) * (1.0f - wx),
                                    fl * (1.0f - wy) * wx);
        ws[2 * t + 1] = make_float4(fl * wy * (1.0f - wx), fl * wy * wx, 0.0f, 0.0f);
    }
    if (t < 2 * CC) {   // partial_cords tail: (C,2) = [x, y] interleaved
        const int cc = t >> 1;
        out[(size_t)CC * FF * HH * WW + t] = (t & 1) ? yin[cc] : xin[cc];
    }
}

// -------- Main kernel: TDM-staged bilinear shift + scale --------
__global__ __launch_bounds__(256) void MoveScalePSF_kernel(
    const float* __restrict__ psf,
    const float4* __restrict__ ws,
    float* __restrict__ out)
{
    __shared__ float smem[(ROWS + 1) * WW];   // 33 KB

    const int tid = threadIdx.x;
    const int cf  = blockIdx.z * FF + blockIdx.y;
    const int i0  = blockIdx.x * ROWS;

    const float4 p0 = ws[2 * cf];
    const float4 p1 = ws[2 * cf + 1];
    const int   ox  = __float_as_int(p0.x);
    const int   oy  = __float_as_int(p0.y);
    const float a00 = p0.z, a01 = p0.w, a10 = p1.x, a11 = p1.y;

    // ---- stage psf rows [i0+oy, i0+oy+ROWS] into LDS ----
    const int sr0 = i0 + oy;
    int fv = sr0 < 0 ? 0 : sr0;                  // first valid source row
    int lv = sr0 + ROWS; if (lv > HH - 1) lv = HH - 1;
    const int nvalid = lv - fv + 1;

    if (nvalid < ROWS + 1) {                     // uniform branch: rare partial/empty tiles
        #pragma unroll 4
        for (int lr = 0; lr <= ROWS; ++lr) {
            const int sr = sr0 + lr;
            if (sr < 0 || sr >= HH) smem[lr * WW + tid] = 0.0f;  // 256 thr cover a row
        }
    }

#if __has_builtin(__builtin_amdgcn_tensor_load_to_lds) && __has_builtin(__builtin_amdgcn_s_wait_tensorcnt)
    if (tid < 32) {                              // wave 0 issues the TDM once
        if (nvalid > 0) {
            const unsigned lds_base = (unsigned)(unsigned long long)(const void*)smem;
            const unsigned lds_addr = lds_base + (unsigned)((fv - sr0) * (WW * 4));
            const unsigned long long ga = (unsigned long long)(const void*)(psf + (size_t)fv * WW);
            const int nv = __builtin_amdgcn_readfirstlane(nvalid);
            u32x4 g0;
            g0[0] = 1u;                                                        // count=1, user desc
            g0[1] = (unsigned)__builtin_amdgcn_readfirstlane((int)lds_addr);   // lds_addr
            g0[2] = (unsigned)__builtin_amdgcn_readfirstlane((int)(ga & 0xFFFFFFFFull));
            g0[3] = ((unsigned)__builtin_amdgcn_readfirstlane((int)((ga >> 32) & 0x1FFFFFFull)))
                    | 0x80000000u;                                             // addr[56:32] | type=2
            i32x8 g1;
            g1[0] = 0x00020000;          // workgroup_mask=0, data_size=2 (4B)
            g1[1] = (int)(WW << 16);     // tensor_dim0[15:0] in [63:48]
            g1[2] = nv << 16;            // tensor_dim1[15:0] in [95:80]
            g1[3] = (int)(WW << 16);     // tile_dim0 in [127:112]
            g1[4] = nv;                  // tile_dim1 in [143:128]
            g1[5] = WW;                  // tensor_dim0_stride[31:0]
            g1[6] = 0;
            g1[7] = 0;
            const i32x4 gz4 = {0, 0, 0, 0};
            const i32x8 gz8 = {0, 0, 0, 0, 0, 0, 0, 0};
            __builtin_amdgcn_tensor_load_to_lds(g0, g1, gz4, gz4, gz8, 0);
        }
        __builtin_amdgcn_s_wait_tensorcnt(0);
    }
#else
    if (nvalid > 0) {
        #pragma unroll 4
        for (int lr = 0; lr <= ROWS; ++lr) {
            const int sr = sr0 + lr;
            if (sr >= 0 && sr < HH) smem[lr * WW + tid] = psf[(size_t)sr * WW + tid];
        }
    }
#endif
    __syncthreads();

    // ---- compute: each thread walks 8 consecutive rows at a fixed 4-col group ----
    // Row-pair reuse: psf row r feeds output rows r-1 and r -> 45 DS loads / 32 px.
    const int j0 = (tid & 63) << 2;        // 0,4,...,252
    const int R  = (tid >> 6) << 3;        // 0,8,16,24

    const int cbase = j0 + ox;             // leftmost needed source column
    bool um[5];
    #pragma unroll
    for (int t = 0; t < 5; ++t) {
        const int cc = cbase + t;
        um[t] = (cc >= 0) & (cc < WW);     // column validity, k-invariant
    }

    float cur[5], nxt[5];
    {
        const float* rp = &smem[R * WW];
        #pragma unroll
        for (int t = 0; t < 5; ++t) cur[t] = um[t] ? rp[cbase + t] : 0.0f;
    }

    const size_t outbase = ((size_t)cf * HH + i0) * WW + j0;
    #pragma unroll
    for (int k = 0; k < 8; ++k) {
        const float* rq = &smem[(R + k + 1) * WW];
        #pragma unroll
        for (int t = 0; t < 5; ++t) nxt[t] = um[t] ? rq[cbase + t] : 0.0f;
        float vv[4];
        #pragma unroll
        for (int jj = 0; jj < 4; ++jj)
            vv[jj] = a00 * cur[jj] + a01 * cur[jj + 1] + a10 * nxt[jj] + a11 * nxt[jj + 1];
        *(float4*)(out + outbase + (size_t)(R + k) * WW) =
            make_float4(vv[0], vv[1], vv[2], vv[3]);
        #pragma unroll
        for (int t = 0; t < 5; ++t) cur[t] = nxt[t];
    }
}

extern "C" void kernel_launch(void* const* d_in, const int* in_sizes, int n_in,
                              void* d_out, int out_size, void* d_ws, size_t ws_size,
                              hipStream_t stream) {
    const float* psf  = (const float*)d_in[0];
    const float* rot  = (const float*)d_in[1];
    const float* x    = (const float*)d_in[2];
    const float* y    = (const float*)d_in[3];
    const float* flux = (const float*)d_in[4];
    float* out = (float*)d_out;
    float4* ws = (float4*)d_ws;   // 512 * 32 B = 16 KB

    MoveScalePSF_params<<<dim3(2), 256, 0, stream>>>(rot, x, y, flux, out, ws);
    dim3 grid(HH / ROWS, FF, CC);   // 8 row-tiles × 32 frames × 16 sources = 4096 blocks
    MoveScalePSF_kernel<<<grid, 256, 0, stream>>>(psf, ws, out);
}